// JilesAthertonStatic_75325136437760
// MI455X (gfx1250) — compile-verified
//
#include <hip/hip_runtime.h>
#include <math.h>

// Jiles-Atherton hysteresis scan for MI455X (gfx1250, wave32).
// T=2048 sequential steps over B=4096 independent columns.
// Latency-bound serial recurrence: the only lever is shrinking the per-step
// dependency chain (hardware v_tanh_f32 / v_rcp_f32, FMA) and wave placement
// (2 waves per SIMD32 to overlap the trans-op latency).

#define TAU_F      6.25e-8f            // 1/16e6
#define INV_TAU_F  1.6e7f
#define MU0_F      1.2566370614359173e-6f
#define INV_MU0_F  7.957747154594767e5f
#define CLIP_F     1.0e8f
#define INV_CLIP_F 1.0e-8f

__device__ __forceinline__ float fast_tanh(float x) {
#if __has_builtin(__builtin_amdgcn_tanhf)
  return __builtin_amdgcn_tanhf(x);        // v_tanh_f32 (gfx1250 trans op)
#else
  float r;
  asm("v_tanh_f32 %0, %1" : "=v"(r) : "v"(x));  // guaranteed hardware tanh
  return r;
#endif
}

__device__ __forceinline__ float fast_rcp(float x) {
  return __builtin_amdgcn_rcpf(x);         // v_rcp_f32
}

__device__ __forceinline__ float sgn(float x) {
  return (float)((x > 0.0f) - (x < 0.0f)); // jnp.sign semantics (0 -> 0)
}

__device__ __forceinline__ float sigmoidf_dev(float x) {
  return 1.0f / (1.0f + __expf(-x));
}

__global__ __launch_bounds__(256) void ja_scan_kernel(
    const float* __restrict__ H0,
    const float* __restrict__ Bseq,
    const float* __restrict__ Msp,
    const float* __restrict__ ap,
    const float* __restrict__ alphap,
    const float* __restrict__ kp,
    const float* __restrict__ cp,
    float* __restrict__ out,
    int T, int B)
{
  const int col = blockIdx.x * blockDim.x + threadIdx.x;
  if (col >= B) return;

  // Squashed physical parameters (wave-uniform; compiler scalarizes these
  // onto the SALU trans path: v_s_exp_f32).
  const float a       = 2000.0f    * sigmoidf_dev(ap[0]);
  const float alpha   = 0.01f      * sigmoidf_dev(alphap[0]);
  const float k       = 1000.0f    * sigmoidf_dev(kp[0]);
  const float c       =              sigmoidf_dev(cp[0]);
  const float Ms      = 6000000.0f * sigmoidf_dev(Msp[0]);
  const float inv_a   = 1.0f / a;
  const float Ms_ov_a = Ms * inv_a;
  const float ck      = c * k;

  float H = H0[col];
  out[col] = H;                              // row 0 = H0 passthrough

  float Bc = Bseq[col];                      // B_seq[0][col]

#pragma unroll 2
  for (int t = 0; t < T - 1; ++t) {
    const size_t nrow = (size_t)(t + 1) * (size_t)B;
    const float Bn = __builtin_nontemporal_load(Bseq + nrow + col);

    // Prefetch ~8 rows ahead (global_prefetch_b8); streamed, read-once data.
    {
      int tp = t + 8; if (tp > T - 1) tp = T - 1;
      __builtin_prefetch(Bseq + (size_t)tp * (size_t)B + col, 0, 0);
    }

    const float dB  = (Bn - Bc) * INV_TAU_F;
    const float M   = __fmaf_rn(Bc, INV_MU0_F, -H);
    const float He  = __fmaf_rn(alpha, M, H);

    // coth_stable(He / a)
    float x = He * inv_a;
    if (fabsf(x) < 1e-7f) x = 1e-7f * sgn(x);
    const float cth  = fast_rcp(fast_tanh(x));
    const float a_He = a * fast_rcp(He);

    const float Man  = Ms * (cth - a_He);
    // dM_an/dH_e = Ms/a * (1 - cth^2 + (a/He)^2)
    const float dMan = Ms_ov_a *
        __fmaf_rn(a_He, a_He, __fmaf_rn(-cth, cth, 1.0f));

    const float diff    = Man - M;
    const float delta   = sgn(dB);
    const float delta_m = 0.5f * (1.0f + sgn(diff * dB));

    const float num  = __fmaf_rn(delta_m, diff, ck * delta * dMan);
    const float den  = __fmaf_rn(-alpha, num, k * delta);
    const float dMdH = num * fast_rcp(den);
    const float dMdB = dMdH * fast_rcp(MU0_F * (1.0f + dMdH));
    const float dMdt = dMdB * dB;

    // softclip(dB/MU0 - dM/dt)
    const float z    = __fmaf_rn(dB, INV_MU0_F, -dMdt);
    const float dHdt = CLIP_F * fast_tanh(z * INV_CLIP_F);

    H = __fmaf_rn(TAU_F, dHdt, H);

    __builtin_nontemporal_store(H, out + nrow + col);  // write-once stream
    Bc = Bn;
  }
}

extern "C" void kernel_launch(void* const* d_in, const int* in_sizes, int n_in,
                              void* d_out, int out_size, void* d_ws, size_t ws_size,
                              hipStream_t stream) {
  // setup_inputs() order: H0, B_seq, Ms_param, a_param, alpha_param, k_param, c_param
  const float* H0     = (const float*)d_in[0];
  const float* Bseq   = (const float*)d_in[1];
  const float* Msp    = (const float*)d_in[2];
  const float* ap     = (const float*)d_in[3];
  const float* alphap = (const float*)d_in[4];
  const float* kp     = (const float*)d_in[5];
  const float* cp     = (const float*)d_in[6];
  float* out = (float*)d_out;

  const int B = in_sizes[0];          // 4096 columns
  const int T = in_sizes[1] / B;      // 2048 time steps

  // 4096 threads = 128 wave32s; 16 blocks x 256 -> 8 waves/WGP on 16 WGPs
  // = 2 waves per SIMD32: enough to hide the trans-op chain latency while
  // staying below the issue-saturation point of the serial recurrence.
  dim3 block(256);
  dim3 grid((B + 255) / 256);
  hipLaunchKernelGGL(ja_scan_kernel, grid, block, 0, stream,
                     H0, Bseq, Msp, ap, alphap, kp, cp, out, T, B);
}